// LightGCN_13503377179279
// MI455X (gfx1250) — compile-verified
//
#include <hip/hip_runtime.h>
#include <hip/hip_bf16.h>

#define NUM_NODES 200000
#define EMB 64

typedef __attribute__((ext_vector_type(2))) float v2f;
typedef __attribute__((ext_vector_type(8))) float v8f;

// AS3 pointer value == byte offset within the wave's LDS allocation.
#define LDS_OFF(p) ((unsigned)(unsigned long long)(__attribute__((address_space(3))) const void*)(const void*)(p))

// One async 4B copy per lane: global -> LDS, tracked with ASYNCcnt.
__device__ __forceinline__ void lgcn_async_b32(unsigned lds_off, const void* g) {
    asm volatile("global_load_async_to_lds_b32 %0, %1, off"
                 :: "v"(lds_off), "v"((unsigned long long)(size_t)g)
                 : "memory");
}

// Stage one 32-edge metadata batch (rows|cols|vals = 128B each) into LDS.
__device__ __forceinline__ void lgcn_stage(unsigned lds_base,
                                           const int* rows, const int* cols,
                                           const float* vals, int base, int lane, int nE) {
    int e = base + lane; if (e > nE - 1) e = nE - 1;   // clamp tail (never consumed)
    lgcn_async_b32(lds_base +       lane * 4, rows + e);
    lgcn_async_b32(lds_base + 128 + lane * 4, cols + e);
    lgcn_async_b32(lds_base + 256 + lane * 4, vals + e);
}

// ---------------------------------------------------------------------------
// init: acc = embeds, x_cur = embeds, x_next = 0   (float4 vectorized)
// ---------------------------------------------------------------------------
__global__ void lgcn_init(const float* __restrict__ e, float* __restrict__ acc,
                          float* __restrict__ xc, float* __restrict__ xn, int n4) {
    int i = blockIdx.x * blockDim.x + threadIdx.x;
    if (i < n4) {
        float4 v = reinterpret_cast<const float4*>(e)[i];
        reinterpret_cast<float4*>(acc)[i] = v;
        reinterpret_cast<float4*>(xc)[i]  = v;
        reinterpret_cast<float4*>(xn)[i]  = make_float4(0.f, 0.f, 0.f, 0.f);
    }
}

// ---------------------------------------------------------------------------
// spmm: x_next[r] += v * x_cur[c]  (COO, one edge per wave iteration)
// Metadata: double-buffered async copy to LDS (ASYNCcnt pipeline), then
// v_readlane broadcast. Row data: 32 lanes x float2 = coalesced 256B gather
// (L2-resident: 51.2MB table << 192MB L2); scatter via global_atomic_add_f32.
// ---------------------------------------------------------------------------
__global__ void lgcn_spmm(const int* __restrict__ rows, const int* __restrict__ cols,
                          const float* __restrict__ vals,
                          const float* __restrict__ x, float* __restrict__ xn, int nE) {
    __shared__ __align__(16) unsigned char smem[8 * 2 * 384];  // 8 waves x 2 bufs x 384B

    const int lane   = threadIdx.x & 31;
    const int wslot  = threadIdx.x >> 5;
    const int wave   = blockIdx.x * (blockDim.x >> 5) + wslot;
    const int nWaves = gridDim.x * (blockDim.x >> 5);
    const int step   = nWaves * 32;

    const unsigned lds0 = LDS_OFF(&smem[wslot * 768]);
    const unsigned bufoff[2] = { lds0, lds0 + 384u };

    int base = wave * 32;
    if (base < nE)
        lgcn_stage(bufoff[0], rows, cols, vals, base, lane, nE);

    int buf = 0;
    for (; base < nE; base += step) {
        const int nb = base + step;
        if (nb < nE) {
            // prefetch next batch, then wait for the current one only
            lgcn_stage(bufoff[buf ^ 1], rows, cols, vals, nb, lane, nE);
            asm volatile("s_wait_asynccnt 3" ::: "memory");
        } else {
            asm volatile("s_wait_asynccnt 0" ::: "memory");
        }

        const unsigned char* mb = &smem[wslot * 768 + (buf ? 384 : 0)];
        int   r = ((const int*)  (mb      ))[lane];   // ds_load_b32, conflict-free
        int   c = ((const int*)  (mb + 128))[lane];
        float v = ((const float*)(mb + 256))[lane];

        const int n = (nE - base) < 32 ? (nE - base) : 32;
#pragma unroll 4
        for (int i = 0; i < n; ++i) {
            int   ri = __builtin_amdgcn_readlane(r, i);
            int   ci = __builtin_amdgcn_readlane(c, i);
            float vi = __int_as_float(__builtin_amdgcn_readlane(__float_as_int(v), i));
            const float2 xv = *reinterpret_cast<const float2*>(x + ci * EMB + lane * 2);
            atomicAdd(xn + ri * EMB + lane * 2,     vi * xv.x);
            atomicAdd(xn + ri * EMB + lane * 2 + 1, vi * xv.y);
        }
        buf ^= 1;
    }
}

// ---------------------------------------------------------------------------
// accum: acc_tile = (s*I) x X_tile + C  via chained V_WMMA_F32_16X16X4_F32.
// One wave per 16x16 tile; EXEC all ones (exact grid, uniform branches only).
//   A(16x4):  lanes 0-15 M=lane, K={0,1}; lanes 16-31 M=lane-16, K={2,3}
//   B(4x16):  VGPR j holds K = 2*half + j, N = lane%16
//   C/D(16x16): VGPR r holds M = r + 8*half, N = lane%16
// final_pass: acc = 0.2*acc + (0.2*I)*X  ==  (acc + x4)/5
// Non-final: also zero zbuf tile (old x_cur -> becomes next x_next).
// ---------------------------------------------------------------------------
__global__ void lgcn_accum_wmma(const float* __restrict__ xnew, float* __restrict__ acc,
                                float* __restrict__ zbuf, int final_pass) {
    const int wave = blockIdx.x * (blockDim.x >> 5) + (threadIdx.x >> 5);
    const int lane = threadIdx.x & 31;
    const int tile_n = wave & 3;          // 64 cols / 16
    const int tile_m = wave >> 2;         // NUM_NODES / 16 tiles down
    const int rb = tile_m * 16;
    const int hf = lane >> 4;
    const int lm = lane & 15;
    const int cn = tile_n * 16 + lm;

    const float s = final_pass ? 0.2f : 1.0f;

    v8f cacc;
#pragma unroll
    for (int r = 0; r < 8; ++r)
        cacc[r] = acc[(rb + r + 8 * hf) * EMB + cn];
    if (final_pass) {
#pragma unroll
        for (int r = 0; r < 8; ++r) cacc[r] *= 0.2f;
    }

#pragma unroll
    for (int kb = 0; kb < 4; ++kb) {
        const int k0 = kb * 4 + 2 * hf;   // global K of this lane-half's first slot
        v2f a, b;
        a.x = (lm == k0)     ? s : 0.0f;  // (s*I)[M=lm, K=k0]
        a.y = (lm == k0 + 1) ? s : 0.0f;
        b.x = xnew[(rb + k0)     * EMB + cn];
        b.y = xnew[(rb + k0 + 1) * EMB + cn];
        cacc = __builtin_amdgcn_wmma_f32_16x16x4_f32(
            /*neg_a=*/false, a, /*neg_b=*/false, b,
            /*c_mod=*/(short)0, cacc, /*reuse_a=*/false, /*reuse_b=*/false);
    }

#pragma unroll
    for (int r = 0; r < 8; ++r)
        acc[(rb + r + 8 * hf) * EMB + cn] = cacc[r];

    if (!final_pass) {
#pragma unroll
        for (int r = 0; r < 8; ++r)
            zbuf[(rb + r + 8 * hf) * EMB + cn] = 0.0f;
    }
}

// ---------------------------------------------------------------------------
extern "C" void kernel_launch(void* const* d_in, const int* in_sizes, int n_in,
                              void* d_out, int out_size, void* d_ws, size_t ws_size,
                              hipStream_t stream) {
    const float* embeds = (const float*)d_in[0];
    const int*   rows   = (const int*)d_in[1];   // jnp.int64 downcast to int32 (x64 off)
    const int*   cols   = (const int*)d_in[2];
    const float* vals   = (const float*)d_in[3];
    float* acc = (float*)d_out;

    float* x0 = (float*)d_ws;
    float* x1 = x0 + (size_t)NUM_NODES * EMB;
    const int nE = in_sizes[1];

    const int n4 = NUM_NODES * EMB / 4;                 // 3.2M float4
    lgcn_init<<<(n4 + 255) / 256, 256, 0, stream>>>(embeds, acc, x0, x1, n4);

    const int spmm_blocks  = 8192;                      // 65536 waves, grid-stride
    const int accum_blocks = NUM_NODES / 32;            // 50000 tiles / 8 waves per block

    float* xc = x0;
    float* xn = x1;
    for (int L = 0; L < 4; ++L) {
        lgcn_spmm<<<spmm_blocks, 256, 0, stream>>>(rows, cols, vals, xc, xn, nE);
        lgcn_accum_wmma<<<accum_blocks, 256, 0, stream>>>(xn, acc, xc, (L == 3) ? 1 : 0);
        float* t = xc; xc = xn; xn = t;                 // zeroed old xc becomes next xn
    }
}